// SpatialGather_Module_5669356833568
// MI455X (gfx1250) — compile-verified
//
#include <hip/hip_runtime.h>
#include <hip/hip_bf16.h>

typedef float v2f __attribute__((ext_vector_type(2)));
typedef float v8f __attribute__((ext_vector_type(8)));

#define B_      8
#define K_      19
#define C_      512
#define HW_     16384
#define CTPW    4                    // c-tiles (16 ch each) per wave
#define SCHUNKS 32
#define SCHUNK  (HW_ / SCHUNKS)      // 512 spatial elements per wave

// ---------------------------------------------------------------------------
// Phase 0: zero the output (atomics accumulate into it; harness poisons it).
// ---------------------------------------------------------------------------
__global__ void zero_out_kernel(float* __restrict__ out, int n) {
  int i = blockIdx.x * blockDim.x + threadIdx.x;
  if (i < n) out[i] = 0.0f;
}

// ---------------------------------------------------------------------------
// Phase 1: per (b,k) spatial softmax stats: row max and 1/sum(exp(x-max)).
// 152 rows of 16384 floats -> ~10 MiB read, negligible vs 256 MiB feats.
// ---------------------------------------------------------------------------
__global__ void softmax_stats_kernel(const float* __restrict__ probs,
                                     float* __restrict__ stats) {
  __shared__ float red[256];
  const int row = blockIdx.x;                 // 0 .. B_*K_-1
  const int tid = threadIdx.x;
  const float* p = probs + (size_t)row * HW_;

  float m = -INFINITY;
  for (int s = tid; s < HW_; s += 256) m = fmaxf(m, p[s]);
  red[tid] = m;
  __syncthreads();
  for (int w = 128; w > 0; w >>= 1) {
    if (tid < w) red[tid] = fmaxf(red[tid], red[tid + w]);
    __syncthreads();
  }
  m = red[0];
  __syncthreads();

  float sum = 0.0f;
  for (int s = tid; s < HW_; s += 256) sum += __expf(p[s] - m);
  red[tid] = sum;
  __syncthreads();
  for (int w = 128; w > 0; w >>= 1) {
    if (tid < w) red[tid] += red[tid + w];
    __syncthreads();
  }
  if (tid == 0) {
    stats[row * 2 + 0] = m;
    stats[row * 2 + 1] = 1.0f / red[0];
  }
}

// ---------------------------------------------------------------------------
// Phase 2: ctx[b,k,c] = sum_s w[b,k,s] * f[b,c,s] via V_WMMA_F32_16X16X4_F32.
//   A (16x4)  = feats tile: M = 16 channels, K = 4 spatial positions
//               lanes 0-15: M=lane, K={0,1} in VGPR{0,1}
//               lanes 16-31: M=lane-16, K={2,3} in VGPR{0,1}
//   B (4x16)  = softmax weights: N = 16 k-columns across lanes 0-15 / 16-31,
//               rows s0+{0,1} / s0+{2,3} per VGPR (mirrors A's lane grouping)
//   D (16x16) = (c_local, k_local) tile, fp32-accumulated over the s-chunk,
//               scattered with global_atomic_add_f32.
// Each wave owns CTPW=4 c-tiles: the softmax B-operand (incl. v_exp_f32) is
// computed ONCE per 4-s step and reused by 8 WMMAs against 4 A-streams ->
// 4x less transcendental work and 4x less probs re-read per feats byte, and
// 8 independent accumulator chains to pipeline the matrix unit.
// One wave per workgroup -> EXEC all-1s for WMMA (required).
// ---------------------------------------------------------------------------
__global__ void __launch_bounds__(32)
ctx_wmma_kernel(const float* __restrict__ feats,
                const float* __restrict__ probs,
                const float* __restrict__ stats,
                float* __restrict__ out) {
  const int sc   = blockIdx.x;          // spatial chunk
  const int cg   = blockIdx.y;          // group of CTPW c-tiles (64 channels)
  const int b    = blockIdx.z;          // batch
  const int lane = threadIdx.x;         // 0..31
  const int m    = lane & 15;           // row within 16-lane group
  const int sgrp = (lane >> 4) << 1;    // s sub-offset for this lane: 0 or 2
  const int cbase = cg * (16 * CTPW);

  // A-operand streams: one channel row per lane per tile, float2 per step.
  // Single base pointer; per-tile deltas become 24-bit immediate offsets.
  const float* fbase =
      feats + (((size_t)b * C_ + cbase + m) * (size_t)HW_) + sgrp;

  // B-operand: k column per lane. Tile1 k=0..15, tile2 k=16..18 (zero-padded).
  const int k1  = m;
  const int k2  = 16 + m;
  const int k2c = (k2 < K_) ? k2 : (K_ - 1);      // clamp address, zero weight
  const float* p1 = probs + (((size_t)b * K_ + k1)  * (size_t)HW_) + sgrp;
  const float* p2 = probs + (((size_t)b * K_ + k2c) * (size_t)HW_) + sgrp;

  const float max1 = stats[(b * K_ + k1) * 2 + 0];
  const float inv1 = stats[(b * K_ + k1) * 2 + 1];
  const float max2 = stats[(b * K_ + k2c) * 2 + 0];
  const float inv2 = (k2 < K_) ? stats[(b * K_ + k2c) * 2 + 1] : 0.0f;

  v8f acc1[CTPW];
  v8f acc2[CTPW];
#pragma unroll
  for (int t = 0; t < CTPW; ++t) {
    acc1[t] = {};
    acc2[t] = {};
  }

  const int s0 = sc * SCHUNK;
#pragma unroll 2
  for (int s = s0; s < s0 + SCHUNK; s += 4) {
    float2 fa[CTPW];
#pragma unroll
    for (int t = 0; t < CTPW; ++t) {
      fa[t] = *(const float2*)(fbase + (size_t)(16 * t) * HW_ + s);
    }
    const float2 q1 = *(const float2*)(p1 + s);
    const float2 q2 = *(const float2*)(p2 + s);

    v2f Bm1;
    Bm1.x = __expf(q1.x - max1) * inv1;
    Bm1.y = __expf(q1.y - max1) * inv1;
    v2f Bm2;
    Bm2.x = __expf(q2.x - max2) * inv2;
    Bm2.y = __expf(q2.y - max2) * inv2;

#pragma unroll
    for (int t = 0; t < CTPW; ++t) {
      v2f A;
      A.x = fa[t].x;
      A.y = fa[t].y;
      // 8 args: (neg_a, A, neg_b, B, c_mod, C, reuse_a, reuse_b)
      acc1[t] = __builtin_amdgcn_wmma_f32_16x16x4_f32(
          false, A, false, Bm1, (short)0, acc1[t], false, false);
      acc2[t] = __builtin_amdgcn_wmma_f32_16x16x4_f32(
          false, A, false, Bm2, (short)0, acc2[t], false, false);
    }
  }

  // D layout: VGPR v, lanes 0-15 -> row M=v; lanes 16-31 -> row M=v+8; N=lane&15
  const int crow = (lane >> 4) * 8;
  float* obase = out + ((size_t)b * C_ + cbase) * (size_t)K_;

  // Unconditional scatter for k-tile 1 (k = 0..15).
#pragma unroll
  for (int t = 0; t < CTPW; ++t) {
#pragma unroll
    for (int v = 0; v < 8; ++v) {
      const int c_local = 16 * t + v + crow;
      unsafeAtomicAdd(obase + (size_t)c_local * K_ + k1, acc1[t][v]);
    }
  }

  // k-tile 2 (k = 16..18): single divergence point around the whole block ->
  // one EXEC save/restore instead of one per atomic.
  if (k2 < K_) {
#pragma unroll
    for (int t = 0; t < CTPW; ++t) {
#pragma unroll
      for (int v = 0; v < 8; ++v) {
        const int c_local = 16 * t + v + crow;
        unsafeAtomicAdd(obase + (size_t)c_local * K_ + k2, acc2[t][v]);
      }
    }
  }
}

// ---------------------------------------------------------------------------
extern "C" void kernel_launch(void* const* d_in, const int* in_sizes, int n_in,
                              void* d_out, int out_size, void* d_ws, size_t ws_size,
                              hipStream_t stream) {
  const float* feats = (const float*)d_in[0];   // (8, 512, 128, 128) fp32
  const float* probs = (const float*)d_in[1];   // (8, 19, 128, 128)  fp32
  float* out   = (float*)d_out;                 // (8, 512, 19, 1)    fp32
  float* stats = (float*)d_ws;                  // 152 * {max, 1/sum} = 1216 B

  zero_out_kernel<<<(out_size + 255) / 256, 256, 0, stream>>>(out, out_size);
  softmax_stats_kernel<<<B_ * K_, 256, 0, stream>>>(probs, stats);

  dim3 grid(SCHUNKS, C_ / (16 * CTPW), B_);     // 32 x 8 x 8 = 2048 waves
  ctx_wmma_kernel<<<grid, 32, 0, stream>>>(feats, probs, stats, out);
}